// MultiHeadClassifier_33827162423889
// MI455X (gfx1250) — compile-verified
//
#include <hip/hip_runtime.h>
#include <hip/hip_bf16.h>

typedef __attribute__((ext_vector_type(2))) float v2f;
typedef __attribute__((ext_vector_type(8))) float v8f;

#define N_SAMP   65536
#define IN_D     128
#define NUM_CAT  16
#define CD       2048            // NUM_CAT * IN_D
#define SEGS     6
#define OUT_W    54              // MAX_SEG + 1
#define EPS      1e-5f
#define SLOPE    0.2f

#define CHUNKS   32
#define ROWS_PER_CHUNK (N_SAMP / CHUNKS)   // 2048
#define SUB_ROWS 64
#define NSUB     (ROWS_PER_CHUNK / SUB_ROWS)  // 32
#define LDS_STRIDE (IN_D + 8)    // 136 floats; row = 544 B (16B aligned), half-waves hit disjoint banks
#define PERM_N   (N_SAMP + 16 * NUM_CAT)   // padded permutation upper bound

__constant__ int d_SHIFTS[NUM_CAT]  = {0,4,6,8,12,16,19,22,24,28,34,36,42,45,48,51};
__constant__ int d_SEGLENS[NUM_CAT] = {4,2,2,4,4,3,3,2,4,6,2,6,3,3,3,3};

// ---- CDNA5 async memory->LDS path (ASYNCcnt), inline asm per CDNA5 ISA
__device__ __forceinline__ uint32_t lds_off(const void* p) {
    // generic pointer to a __shared__ object keeps the LDS byte offset in addr[31:0]
    return (uint32_t)(uintptr_t)p;
}
__device__ __forceinline__ void async_b128(uint32_t ldsOff, const float* gsrc) {
    asm volatile("global_load_async_to_lds_b128 %0, %1, off"
                 :: "v"(ldsOff), "v"(gsrc) : "memory");
}
__device__ __forceinline__ void wait_async0() {
    asm volatile("s_wait_asynccnt 0x0" ::: "memory");
}
__device__ __forceinline__ void wait_async8() {
    asm volatile("s_wait_asynccnt 0x8" ::: "memory");
}

// ---------------------------------------------------------------- init
__global__ void init_kernel(int* counts, int* cursors, int* perm) {
    int i = blockIdx.x * blockDim.x + threadIdx.x;
    if (i < NUM_CAT) { counts[i] = 0; cursors[i] = 0; }
    if (i < PERM_N) perm[i] = -1;
}

// ---------------------------------------------------- column sums of F
__global__ __launch_bounds__(128)
void colsum_part_kernel(const float* __restrict__ F, float* __restrict__ part) {
    int b = blockIdx.x, d = threadIdx.x;
    const float* Fb = F + (size_t)b * ROWS_PER_CHUNK * IN_D;
    float s = 0.f;
    for (int r = 0; r < ROWS_PER_CHUNK; ++r) s += Fb[(size_t)r * IN_D + d];
    part[b * IN_D + d] = s;
}

__global__ __launch_bounds__(128)
void colsum_red_kernel(const float* __restrict__ part, float* __restrict__ csum) {
    int d = threadIdx.x;
    float s = 0.f;
    for (int c = 0; c < CHUNKS; ++c) s += part[c * IN_D + d];   // fixed order
    csum[d] = s;
}

// ------------- Gram G = F^T F via WMMA f32, async double-buffered LDS
__global__ __launch_bounds__(256)
void gram_kernel(const float* __restrict__ F, float* __restrict__ gpart) {
    __shared__ float Fs[2][SUB_ROWS][LDS_STRIDE];
    const int tid  = threadIdx.x;
    const int wave = tid >> 5, lane = tid & 31;
    const int ln15 = lane & 15, lhi = lane >> 4;
    const int chunk = blockIdx.x;

    v8f acc[8];
    for (int t = 0; t < 8; ++t) acc[t] = (v8f){0.f,0.f,0.f,0.f,0.f,0.f,0.f,0.f};

    const float* Fbase = F + (size_t)chunk * ROWS_PER_CHUNK * IN_D;
    const uint32_t fsBase[2] = { lds_off(&Fs[0][0][0]), lds_off(&Fs[1][0][0]) };

    // each thread owns 8 b128 chunks of a 64x128 sub-tile
    auto issue = [&](int sub) {
        const float* src = Fbase + (size_t)sub * SUB_ROWS * IN_D;
        uint32_t base = fsBase[sub & 1];
        for (int q = 0; q < 8; ++q) {
            int p4 = tid + q * 256;          // float4 index in sub-tile
            int r = p4 >> 5, e4 = p4 & 31;   // row, float4-within-row
            async_b128(base + (uint32_t)((r * LDS_STRIDE + e4 * 4) * 4),
                       src + (size_t)r * IN_D + e4 * 4);
        }
    };

    issue(0);
    issue(1);
    for (int sub = 0; sub < NSUB; ++sub) {
        if (sub < NSUB - 1) wait_async8();   // current buffer's 8 chunks retired (in-order)
        else                wait_async0();
        __syncthreads();
        const float (*Fb)[LDS_STRIDE] = Fs[sub & 1];
        for (int k0 = 0; k0 < SUB_ROWS; k0 += 4) {
            int kb = k0 + 2 * lhi;
            v2f b; b.x = Fb[kb][wave * 16 + ln15]; b.y = Fb[kb + 1][wave * 16 + ln15];
            for (int tm = 0; tm < 8; ++tm) {
                v2f a; a.x = Fb[kb][tm * 16 + ln15]; a.y = Fb[kb + 1][tm * 16 + ln15];
                acc[tm] = __builtin_amdgcn_wmma_f32_16x16x4_f32(
                    false, a, false, b, (short)0, acc[tm], false, false);
            }
        }
        __syncthreads();                      // buffer free for reuse
        if (sub + 2 < NSUB) issue(sub + 2);   // prefetch into the just-released buffer
    }

    float* gp = gpart + (size_t)chunk * IN_D * IN_D;
    for (int tm = 0; tm < 8; ++tm)
        for (int r = 0; r < 8; ++r) {
            int gm = tm * 16 + r + 8 * lhi;
            int gn = wave * 16 + ln15;
            gp[gm * IN_D + gn] = acc[tm][r];
        }
}

__global__ __launch_bounds__(256)
void gram_red_kernel(const float* __restrict__ gpart, float* __restrict__ G) {
    int i = blockIdx.x * blockDim.x + threadIdx.x;
    if (i >= IN_D * IN_D) return;
    float s = 0.f;
    for (int c = 0; c < CHUNKS; ++c) s += gpart[(size_t)c * IN_D * IN_D + i];  // fixed order
    G[i] = s;
}

// ------------------- BN fold: scale/shift per column from G and colsum
__global__ __launch_bounds__(128)
void bnparams_kernel(const float* __restrict__ W1, const float* __restrict__ gamma,
                     const float* __restrict__ beta, const float* __restrict__ G,
                     const float* __restrict__ csum,
                     float* __restrict__ scale, float* __restrict__ shift) {
    __shared__ float w[IN_D];
    __shared__ float red[IN_D];
    __shared__ float red2[IN_D];
    const int j = blockIdx.x, d = threadIdx.x;
    w[d] = W1[(size_t)d * CD + j];
    __syncthreads();
    float t = 0.f;
    const float* Gr = G + d * IN_D;
    for (int e = 0; e < IN_D; ++e) t += Gr[e] * w[e];
    red[d]  = t * w[d];        // contribution to w^T G w
    red2[d] = csum[d] * w[d];  // contribution to colsum . w
    __syncthreads();
    for (int s = 64; s > 0; s >>= 1) {
        if (d < s) { red[d] += red[d + s]; red2[d] += red2[d + s]; }
        __syncthreads();
    }
    if (d == 0) {
        const float invN = 1.0f / (float)N_SAMP;
        float mean = red2[0] * invN;
        float var  = red[0] * invN - mean * mean;     // biased, as torch BN
        float istd = rsqrtf(var + EPS);
        float sc = gamma[j] * istd;
        scale[j] = sc;
        shift[j] = beta[j] - mean * sc;
    }
}

// ----------------------------------------- category bucketing (padded)
__global__ void hist_kernel(const int* __restrict__ labels, int* __restrict__ counts) {
    int n = blockIdx.x * blockDim.x + threadIdx.x;
    if (n < N_SAMP) atomicAdd(&counts[labels[n]], 1);
}

__global__ void offsets_kernel(const int* __restrict__ counts, int* __restrict__ offs) {
    if (threadIdx.x == 0 && blockIdx.x == 0) {
        int acc = 0;
        for (int c = 0; c < NUM_CAT; ++c) {
            offs[c] = acc;
            acc += ((counts[c] + 15) / 16) * 16;    // pad each group to tile size
        }
        offs[NUM_CAT] = acc;
    }
}

__global__ void scatter_kernel(const int* __restrict__ labels, const int* __restrict__ offs,
                               int* __restrict__ cursors, int* __restrict__ perm) {
    int n = blockIdx.x * blockDim.x + threadIdx.x;
    if (n >= N_SAMP) return;
    int c = labels[n];
    int pos = offs[c] + atomicAdd(&cursors[c], 1);
    perm[pos] = n;
}

// -------------- classifier: async-staged WMMA GEMM + BN + log-softmax
__global__ __launch_bounds__(256)
void classify_kernel(const float* __restrict__ F, const float* __restrict__ W1,
                     const float* __restrict__ W2, const float* __restrict__ bias,
                     const float* __restrict__ scale, const float* __restrict__ shift,
                     const int* __restrict__ offs, const int* __restrict__ perm,
                     float* __restrict__ out) {
    __shared__ float Ws[IN_D][LDS_STRIDE];   // W1 category slice, k-major
    __shared__ float Hs[16][LDS_STRIDE];     // A tile, later reused as normalized H
    __shared__ float W2s[IN_D * SEGS];       // 768 floats
    __shared__ int   idx[16];
    __shared__ int   catS;
    __shared__ float lg[16][8];              // 6 logits + lse

    const int tid  = threadIdx.x;
    const int base = blockIdx.x * 16;
    const int total = offs[NUM_CAT];
    if (base >= total) return;               // uniform across block

    if (tid < 16) idx[tid] = perm[base + tid];      // padding slots hold -1
    if (tid == 0) {
        int c = 0;
        while (c < NUM_CAT - 1 && base >= offs[c + 1]) ++c;
        catS = c;
    }
    __syncthreads();
    const int cat = catS;

    // ---- async stage: W1 slice (64KB), W2 slice (3KB), gathered F rows (8KB)
    const uint32_t wsBase  = lds_off(&Ws[0][0]);
    const uint32_t hsBase  = lds_off(&Hs[0][0]);
    const uint32_t w2Base  = lds_off(&W2s[0]);

    const float* w1base = W1 + cat * IN_D;            // row k starts at W1[k*CD + cat*128]
    for (int q = 0; q < 16; ++q) {
        int p4 = tid + q * 256;                       // 0..4095 float4 chunks
        int k = p4 >> 5, e4 = p4 & 31;
        async_b128(wsBase + (uint32_t)((k * LDS_STRIDE + e4 * 4) * 4),
                   w1base + (size_t)k * CD + e4 * 4);
    }
    if (tid < (IN_D * SEGS) / 4)                      // 192 chunks
        async_b128(w2Base + (uint32_t)(tid * 16),
                   W2 + (size_t)cat * IN_D * SEGS + tid * 4);
    for (int q = 0; q < 2; ++q) {
        int p4 = tid + q * 256;                       // 0..511: 16 rows x 32 chunks
        int i = p4 >> 5, e4 = p4 & 31;
        int s = idx[i];
        if (s >= 0) {
            async_b128(hsBase + (uint32_t)((i * LDS_STRIDE + e4 * 4) * 4),
                       F + (size_t)s * IN_D + e4 * 4);
        } else {
            Hs[i][e4 * 4 + 0] = 0.f; Hs[i][e4 * 4 + 1] = 0.f;
            Hs[i][e4 * 4 + 2] = 0.f; Hs[i][e4 * 4 + 3] = 0.f;
        }
    }
    wait_async0();
    __syncthreads();

    const int wave = tid >> 5, lane = tid & 31;
    const int ln15 = lane & 15, lhi = lane >> 4;

    v8f acc = (v8f){0.f,0.f,0.f,0.f,0.f,0.f,0.f,0.f};
    for (int k0 = 0; k0 < IN_D; k0 += 4) {
        int kb = k0 + 2 * lhi;
        v2f a; a.x = Hs[ln15][kb];                 a.y = Hs[ln15][kb + 1];
        v2f b; b.x = Ws[kb][wave * 16 + ln15];     b.y = Ws[kb + 1][wave * 16 + ln15];
        acc = __builtin_amdgcn_wmma_f32_16x16x4_f32(
            false, a, false, b, (short)0, acc, false, false);
    }
    __syncthreads();   // done reading A; safe to overwrite with H

    // BN (folded scale/shift) + LeakyReLU, write back to LDS
    for (int r = 0; r < 8; ++r) {
        int m = r + 8 * lhi;           // sample row in tile
        int n = wave * 16 + ln15;      // d within 128
        int j = cat * IN_D + n;
        float v = acc[r] * scale[j] + shift[j];
        v = (v >= 0.f) ? v : SLOPE * v;
        Hs[m][n] = v;
    }
    __syncthreads();

    // logits[i][s] = bias[s] + H[i,:] . W2[cat,:,s]
    if (tid < 16 * SEGS) {
        int i = tid / SEGS, s = tid % SEGS;
        float a2 = bias[s];
        for (int d = 0; d < IN_D; ++d) a2 += Hs[i][d] * W2s[d * SEGS + s];
        lg[i][s] = a2;
    }
    __syncthreads();

    if (tid < 16) {
        float m = lg[tid][0];
        for (int s = 1; s < SEGS; ++s) m = fmaxf(m, lg[tid][s]);
        float se = 0.f;
        for (int s = 0; s < SEGS; ++s) se += __expf(lg[tid][s] - m);
        lg[tid][6] = m + __logf(se);   // lse
    }
    __syncthreads();

    // scatter full 54-wide rows (zeros outside the segment range)
    for (int p = tid; p < 16 * OUT_W; p += 256) {
        int i = p / OUT_W, col = p % OUT_W;
        int s = idx[i];
        if (s < 0) continue;
        int sh = d_SHIFTS[cat], ln = d_SEGLENS[cat];
        float v = 0.f;
        if (col >= sh && col < sh + ln) v = lg[i][col - sh] - lg[i][6];
        out[(size_t)s * OUT_W + col] = v;
    }
}

// ---------------------------------------------------------------- host
extern "C" void kernel_launch(void* const* d_in, const int* in_sizes, int n_in,
                              void* d_out, int out_size, void* d_ws, size_t ws_size,
                              hipStream_t stream) {
    (void)in_sizes; (void)n_in; (void)out_size; (void)ws_size;
    const float* features = (const float*)d_in[0];
    const float* W1       = (const float*)d_in[1];
    const float* gamma    = (const float*)d_in[2];
    const float* beta     = (const float*)d_in[3];
    const float* W2       = (const float*)d_in[4];
    const float* bias     = (const float*)d_in[5];
    const int*   labels   = (const int*)d_in[6];
    float* out = (float*)d_out;

    // workspace layout (floats then ints), ~2.5 MB total
    float* gpart  = (float*)d_ws;                       // CHUNKS*128*128
    float* G      = gpart + (size_t)CHUNKS * IN_D * IN_D;
    float* cspart = G + IN_D * IN_D;                    // CHUNKS*128
    float* csum   = cspart + CHUNKS * IN_D;             // 128
    float* scale  = csum + IN_D;                        // 2048
    float* shift  = scale + CD;                         // 2048
    int* counts   = (int*)(shift + CD);                 // 16
    int* offs     = counts + NUM_CAT;                   // 17
    int* cursors  = offs + NUM_CAT + 1;                 // 16
    int* perm     = cursors + NUM_CAT;                  // PERM_N

    init_kernel<<<(PERM_N + 255) / 256, 256, 0, stream>>>(counts, cursors, perm);

    colsum_part_kernel<<<CHUNKS, 128, 0, stream>>>(features, cspart);
    colsum_red_kernel<<<1, 128, 0, stream>>>(cspart, csum);

    gram_kernel<<<CHUNKS, 256, 0, stream>>>(features, gpart);
    gram_red_kernel<<<(IN_D * IN_D + 255) / 256, 256, 0, stream>>>(gpart, G);

    bnparams_kernel<<<CD, 128, 0, stream>>>(W1, gamma, beta, G, csum, scale, shift);

    hist_kernel<<<(N_SAMP + 255) / 256, 256, 0, stream>>>(labels, counts);
    offsets_kernel<<<1, 1, 0, stream>>>(counts, offs);
    scatter_kernel<<<(N_SAMP + 255) / 256, 256, 0, stream>>>(labels, offs, cursors, perm);

    const int maxTiles = (N_SAMP + 16 * (NUM_CAT - 1) + 15) / 16 + 1;  // 4112
    classify_kernel<<<maxTiles, 256, 0, stream>>>(features, W1, W2, bias,
                                                  scale, shift, offs, perm, out);
}